// ImageAttnModule_22299470201522
// MI455X (gfx1250) — compile-verified
//
#include <hip/hip_runtime.h>
#include <math.h>

typedef __attribute__((ext_vector_type(2))) float v2f;
typedef __attribute__((ext_vector_type(8))) float v8f;

#define B_ 128
#define S_ 1024
#define D_ 512
#define N_ 784
#define NV4 196                 // N_ / 4, exact

// ---------------------------------------------------------------------------
// K0: fold v = Wa @ Wc  -> v[0:512] = v1 (scales q*img), v[512:1024] = v2 (img)
// ---------------------------------------------------------------------------
__global__ __launch_bounds__(256) void k_fold_wc(const float* __restrict__ Wa,
                                                 const float* __restrict__ Wc,
                                                 float* __restrict__ v) {
    const int j = blockIdx.x * 256 + threadIdx.x;      // 0..1023
    float acc = 0.0f;
    for (int d = 0; d < D_; ++d)
        acc += Wa[d] * Wc[d * (2 * D_) + j];           // coalesced over j
    v[j] = acc;
}

// ---------------------------------------------------------------------------
// K1: query = in_state @ Wq.T + bq, fused into w[b,d] = v1[d]*query + v2[d]
//     WMMA f32 16x16x4: one 16x16 output tile per wave, K-loop over S.
//     A[m][k] = in_state[(tm+m)*S + k], B[k][n] = Wq[(tn+n)*S + k]
// ---------------------------------------------------------------------------
__global__ __launch_bounds__(256) void k_query_w(const float* __restrict__ in_state,
                                                 const float* __restrict__ Wq,
                                                 const float* __restrict__ bq,
                                                 const float* __restrict__ v,
                                                 float* __restrict__ w) {
    const int wave = threadIdx.x >> 5;
    const int lane = threadIdx.x & 31;
    const int tile = blockIdx.x * 8 + wave;            // 256 tiles: 8 (M) x 32 (N)
    const int tm = (tile & 7) << 4;                    // batch-row tile origin
    const int tn = (tile >> 3) << 4;                   // d-col tile origin
    const int r  = lane & 15;                          // M for A, N for B
    const int h  = lane >> 4;                          // K-half selector

    v8f acc = {};
    for (int k0 = 0; k0 < S_; k0 += 4) {
        const int ka = k0 + 2 * h;
        v2f a, b;
        a.x = in_state[(tm + r) * S_ + ka];
        a.y = in_state[(tm + r) * S_ + ka + 1];
        b.x = Wq[(tn + r) * S_ + ka];
        b.y = Wq[(tn + r) * S_ + ka + 1];
        acc = __builtin_amdgcn_wmma_f32_16x16x4_f32(false, a, false, b,
                                                    (short)0, acc, false, false);
    }
#pragma unroll
    for (int j = 0; j < 8; ++j) {
        const int row = tm + j + 8 * h;                // batch index b
        const int col = tn + r;                        // channel index d
        const float q = acc[j] + bq[col];
        w[row * D_ + col] = v[col] * q + v[D_ + col];
    }
}

// ---------------------------------------------------------------------------
// K2: per-batch fused  logits -> softmax -> pooled.
//     One block per batch. image[b] (1.6 MB) is streamed with b128 loads +
//     global_prefetch; second pass re-reads it from L2 with b128 loads and
//     ds_load_b128 of the LDS-resident attention vector.
// ---------------------------------------------------------------------------
__global__ __launch_bounds__(256) void k_attn_pool(const float* __restrict__ image,
                                                   const float* __restrict__ w,
                                                   float* __restrict__ pooled) {
    __shared__ float s_w[D_];
    __shared__ __align__(16) float s_attn[N_];
    __shared__ float s_red[256];

    const int b = blockIdx.x;
    const int t = threadIdx.x;
    const float4* __restrict__ imgv =
        (const float4*)(image + (size_t)b * D_ * N_);  // row stride NV4 float4s

    for (int d = t; d < D_; d += 256) s_w[d] = w[b * D_ + d];
    __syncthreads();

    // ---- pass A: logits[b, 4t..4t+3] = sum_d w[b,d] * image[b,d,4t..4t+3] ----
    const bool act = (t < NV4);                        // 196 active lanes, exact fit
    float4 a4 = make_float4(0.f, 0.f, 0.f, 0.f);
    for (int d = 0; d < D_; ++d) {
        const float wv = s_w[d];
        if (act) {
            if ((d & 15) == 0 && d + 16 < D_)
                __builtin_prefetch(&imgv[(d + 16) * NV4 + t], 0, 1);  // global_prefetch_b8
            const float4 x = imgv[d * NV4 + t];        // global_load_b128
            a4.x += wv * x.x;
            a4.y += wv * x.y;
            a4.z += wv * x.z;
            a4.w += wv * x.w;
        }
    }

    // ---- softmax over n (block reductions; inactive lanes contribute identity) ----
    float m = act ? fmaxf(fmaxf(a4.x, a4.y), fmaxf(a4.z, a4.w)) : -INFINITY;
    s_red[t] = m; __syncthreads();
    for (int s = 128; s > 0; s >>= 1) {
        if (t < s) s_red[t] = fmaxf(s_red[t], s_red[t + s]);
        __syncthreads();
    }
    const float mall = s_red[0]; __syncthreads();

    float4 e4 = make_float4(0.f, 0.f, 0.f, 0.f);
    if (act) {
        e4.x = expf(a4.x - mall);
        e4.y = expf(a4.y - mall);
        e4.z = expf(a4.z - mall);
        e4.w = expf(a4.w - mall);
    }
    s_red[t] = e4.x + e4.y + e4.z + e4.w; __syncthreads();
    for (int s = 128; s > 0; s >>= 1) {
        if (t < s) s_red[t] += s_red[t + s];
        __syncthreads();
    }
    const float inv = 1.0f / s_red[0]; __syncthreads();

    if (act) {
        float4* s_attnv = (float4*)s_attn;
        s_attnv[t] = make_float4(e4.x * inv, e4.y * inv, e4.z * inv, e4.w * inv);
    }
    __syncthreads();

    // ---- pass B: pooled[b,d] = sum_n attn[n] * image[b,d,n] (one d per wave) ----
    const int wave = t >> 5, lane = t & 31;
    const float4* s_attnv = (const float4*)s_attn;
    for (int d = wave; d < D_; d += 8) {
        const float4* __restrict__ row = imgv + d * NV4;
        float s = 0.f;
        for (int c = lane; c < NV4; c += 32) {         // b128 global + b128 LDS
            const float4 x = row[c];
            const float4 a = s_attnv[c];
            s += a.x * x.x + a.y * x.y + a.z * x.z + a.w * x.w;
        }
#pragma unroll
        for (int off = 16; off > 0; off >>= 1) s += __shfl_down(s, off, 32);
        if (lane == 0) pooled[b * D_ + d] = s;
    }
}

// ---------------------------------------------------------------------------
// K3: out = pooled @ Wo.T + bo   (WMMA f32 16x16x4, same scheme as K1)
//     A[m][k] = pooled[(tm+m)*D + k], B[k][n] = Wo[(tn+n)*D + k]
// ---------------------------------------------------------------------------
__global__ __launch_bounds__(256) void k_out_gemm(const float* __restrict__ pooled,
                                                  const float* __restrict__ Wo,
                                                  const float* __restrict__ bo,
                                                  float* __restrict__ out) {
    const int wave = threadIdx.x >> 5;
    const int lane = threadIdx.x & 31;
    const int tile = blockIdx.x * 8 + wave;            // 512 tiles: 8 (M) x 64 (N)
    const int tm = (tile & 7) << 4;                    // batch rows
    const int tn = (tile >> 3) << 4;                   // s cols
    const int r  = lane & 15;
    const int h  = lane >> 4;

    v8f acc = {};
    for (int k0 = 0; k0 < D_; k0 += 4) {
        const int ka = k0 + 2 * h;
        v2f a, b;
        a.x = pooled[(tm + r) * D_ + ka];
        a.y = pooled[(tm + r) * D_ + ka + 1];
        b.x = Wo[(tn + r) * D_ + ka];
        b.y = Wo[(tn + r) * D_ + ka + 1];
        acc = __builtin_amdgcn_wmma_f32_16x16x4_f32(false, a, false, b,
                                                    (short)0, acc, false, false);
    }
#pragma unroll
    for (int j = 0; j < 8; ++j) {
        const int row = tm + j + 8 * h;
        const int col = tn + r;
        out[row * S_ + col] = acc[j] + bo[col];
    }
}

// ---------------------------------------------------------------------------
extern "C" void kernel_launch(void* const* d_in, const int* in_sizes, int n_in,
                              void* d_out, int out_size, void* d_ws, size_t ws_size,
                              hipStream_t stream) {
    const float* in_state = (const float*)d_in[0];
    const float* image    = (const float*)d_in[1];
    const float* Wq       = (const float*)d_in[2];
    const float* bq       = (const float*)d_in[3];
    const float* Wc       = (const float*)d_in[4];
    // d_in[5] = bc : constant shift under softmax -> provably no effect on output
    const float* Wa       = (const float*)d_in[6];
    // d_in[7] = ba : same
    const float* Wo       = (const float*)d_in[8];
    const float* bo       = (const float*)d_in[9];
    float* out = (float*)d_out;

    float* ws     = (float*)d_ws;
    float* v      = ws;                    // 1024 floats
    float* w      = v + 2 * D_;            // B*D floats
    float* pooled = w + B_ * D_;           // B*D floats

    k_fold_wc <<<  4, 256, 0, stream>>>(Wa, Wc, v);
    k_query_w <<< 32, 256, 0, stream>>>(in_state, Wq, bq, v, w);
    k_attn_pool<<<B_, 256, 0, stream>>>(image, w, pooled);
    k_out_gemm<<< 64, 256, 0, stream>>>(pooled, Wo, bo, out);
}